// LSTMCell_82282983457013
// MI455X (gfx1250) — compile-verified
//
#include <hip/hip_runtime.h>

typedef __attribute__((ext_vector_type(2))) float v2f;
typedef __attribute__((ext_vector_type(8))) float v8f;

#define KT   32
#define LDSW (KT + 4)   // padded row stride (36 floats): conflict-free frag reads, 16B-aligned float4 slots

// ---- CDNA5 async global->LDS copy (bypasses VGPRs; tracked by ASYNCcnt) ----
#if __has_builtin(__builtin_amdgcn_global_load_async_to_lds_b128)
typedef int v4i __attribute__((vector_size(16)));
typedef __attribute__((address_space(1))) v4i* v4i_g;   // global (AS1) int4*
typedef __attribute__((address_space(3))) v4i* v4i_l;   // LDS    (AS3) int4*
__device__ __forceinline__ void copy16_to_lds(void* lds_dst, const void* gsrc) {
    __builtin_amdgcn_global_load_async_to_lds_b128((v4i_g)gsrc, (v4i_l)lds_dst, 0, 0);
}
#else
__device__ __forceinline__ void copy16_to_lds(void* lds_dst, const void* gsrc) {
    *(float4*)lds_dst = *(const float4*)gsrc;
}
#endif

__device__ __forceinline__ void wait_async0() {
#if __has_builtin(__builtin_amdgcn_s_wait_asynccnt)
    __builtin_amdgcn_s_wait_asynccnt(0);
#else
    asm volatile("s_wait_asynccnt 0x0" ::: "memory");
#endif
}

__device__ __forceinline__ float fast_sigmoid(float x) {
    return 1.0f / (1.0f + __expf(-x));
}
__device__ __forceinline__ float fast_tanh(float x) {
    // tanh(x) = 2*sigmoid(2x) - 1 ; saturates correctly for large |x|
    return 2.0f / (1.0f + __expf(-2.0f * x)) - 1.0f;
}

// Fused LSTM cell: Z = [X|h] @ W^T + b for 4 gates, then gating epilogue.
// Workgroup: 256 threads = 8 wave32, 2(m) x 4(n). Tile: 32 rows x 64 cols per gate.
// Double-buffered LDS staging via async global->LDS loads.
__global__ __launch_bounds__(256)
void lstm_fused_wmma(const float* __restrict__ X, const float* __restrict__ h,
                     const float* __restrict__ c,
                     const float* __restrict__ Wf, const float* __restrict__ bf,
                     const float* __restrict__ Wi, const float* __restrict__ bi,
                     const float* __restrict__ Wg, const float* __restrict__ bg,
                     const float* __restrict__ Wo, const float* __restrict__ bo,
                     float* __restrict__ outH, float* __restrict__ outC,
                     int N, int B)
{
    const int K2 = 2 * N;

    __shared__ float sA[2][32][LDSW];      // double-buffered: 32 rows of XH, KT cols
    __shared__ float sW[2][4][64][LDSW];   // double-buffered: 4 gates x 64 W-rows x KT cols

    const int tid    = threadIdx.x;
    const int wid    = tid >> 5;
    const int lane   = tid & 31;
    const int wave_m = wid >> 2;           // 0..1
    const int wave_n = wid & 3;            // 0..3

    const int m0 = blockIdx.y * 32;        // batch-row tile
    const int n0 = blockIdx.x * 64;        // per-gate output-col tile

    // gate order: 0=f, 1=g, 2=o, 3=i (matches reference W_all concat)
    const float* Wp[4] = {Wf, Wg, Wo, Wi};

    v8f acc[4];
    #pragma unroll
    for (int g = 0; g < 4; ++g) acc[g] = (v8f){};

    // staging indices: one 16B async copy per thread for A, 8 for W
    const int arow = tid >> 3;             // 0..31
    const int ac4  = (tid & 7) << 2;       // 0,4,...,28

    // WMMA fp32 16x16x4 fragment layout: lanes 0-15 hold K={0,1}, lanes 16-31 hold K={2,3}
    const int mrow = wave_m * 16 + (lane & 15);   // row in sA
    const int nrow = wave_n * 16 + (lane & 15);   // row in sW[g]
    const int kh   = (lane >> 4) << 1;            // 0 or 2

    auto stage = [&](int buf, int kb) {
        const float* Asrc = (kb < N)
            ? (X + (size_t)(m0 + arow) * N + (kb + ac4))
            : (h + (size_t)(m0 + arow) * N + (kb - N + ac4));
        copy16_to_lds(&sA[buf][arow][ac4], Asrc);
        #pragma unroll
        for (int g = 0; g < 4; ++g) {
            #pragma unroll
            for (int r = 0; r < 2; ++r) {
                const int idx  = (r << 8) + tid;
                const int wrow = idx >> 3;           // 0..63
                const int wc4  = (idx & 7) << 2;
                copy16_to_lds(&sW[buf][g][wrow][wc4],
                              Wp[g] + (size_t)(n0 + wrow) * K2 + kb + wc4);
            }
        }
    };

    const int T = K2 / KT;                 // 128 K-tiles
    stage(0, 0);

    for (int t = 0; t < T; ++t) {
        wait_async0();       // this wave's async loads for tile t are in LDS
        __syncthreads();     // ...and every other wave's too (also fences buf reuse)

        if (t + 1 < T) stage((t + 1) & 1, (t + 1) * KT);   // prefetch next tile

        const int buf = t & 1;
        #pragma unroll
        for (int kk = 0; kk < KT; kk += 4) {
            v2f a = *(const v2f*)&sA[buf][mrow][kk + kh];
            #pragma unroll
            for (int g = 0; g < 4; ++g) {
                v2f b = *(const v2f*)&sW[buf][g][nrow][kk + kh];
                acc[g] = __builtin_amdgcn_wmma_f32_16x16x4_f32(
                    false, a, false, b, (short)0, acc[g], false, false);
            }
        }
    }

    // ---- fused LSTM epilogue ----
    // C/D layout: VGPR j, lanes 0-15 -> (M=j, N=lane); lanes 16-31 -> (M=j+8, N=lane-16)
    const int ncl  = lane & 15;
    const int gcol = n0 + wave_n * 16 + ncl;
    const float bfv = bf[gcol];
    const float bgv = bg[gcol];
    const float bov = bo[gcol];
    const float biv = bi[gcol];
    const int mtop = m0 + wave_m * 16 + ((lane >> 4) << 3);

    #pragma unroll
    for (int j = 0; j < 8; ++j) {
        const size_t o = (size_t)(mtop + j) * N + gcol;
        const float zf = acc[0][j] + bfv;
        const float zg = acc[1][j] + bgv;
        const float zo = acc[2][j] + bov;
        const float zi = acc[3][j] + biv;
        const float cn = fast_sigmoid(zf) * c[o] + fast_sigmoid(zg) * fast_tanh(zi);
        const float hn = fast_sigmoid(zo) * fast_tanh(cn);
        outH[o] = hn;
        outC[o] = cn;
    }
}

extern "C" void kernel_launch(void* const* d_in, const int* in_sizes, int n_in,
                              void* d_out, int out_size, void* d_ws, size_t ws_size,
                              hipStream_t stream) {
    (void)n_in; (void)out_size; (void)d_ws; (void)ws_size;

    const float* X  = (const float*)d_in[0];
    const float* h  = (const float*)d_in[1];
    const float* c  = (const float*)d_in[2];
    const float* Wf = (const float*)d_in[3];
    const float* bf = (const float*)d_in[4];
    const float* Wi = (const float*)d_in[5];
    const float* bi = (const float*)d_in[6];
    const float* Wg = (const float*)d_in[7];
    const float* bg = (const float*)d_in[8];
    const float* Wo = (const float*)d_in[9];
    const float* bo = (const float*)d_in[10];

    const int N = in_sizes[4];            // bias length = n_features (2048)
    const int B = in_sizes[0] / N;        // batch (4096)

    float* outH = (float*)d_out;
    float* outC = outH + (size_t)B * N;

    dim3 grid(N / 64, B / 32);
    dim3 block(256);
    lstm_fused_wmma<<<grid, block, 0, stream>>>(
        X, h, c, Wf, bf, Wi, bi, Wg, bg, Wo, bo, outH, outC, N, B);
}